// GCNClassifier_2740189135665
// MI455X (gfx1250) — compile-verified
//
#include <hip/hip_runtime.h>
#include <hip/hip_bf16.h>

typedef __attribute__((ext_vector_type(2))) float v2f;
typedef __attribute__((ext_vector_type(8))) float v8f;

#define NNODES 100000
#define NEDGES 1600000
#define INDIM  518
#define HID    128
#define MLPH   64
#define OUTD   4

// ---------------------------------------------------------------------------
// degree / normalization
// ---------------------------------------------------------------------------
__global__ __launch_bounds__(256) void deg_count(const long long* __restrict__ ei,
                                                 unsigned* __restrict__ deg, int E) {
    int i = blockIdx.x * blockDim.x + threadIdx.x;
    int stride = gridDim.x * blockDim.x;
    for (int e = i; e < E; e += stride) {
        int d = (int)ei[(long)E + e];          // dst row of edge_index
        atomicAdd(&deg[d], 1u);
    }
}

__global__ __launch_bounds__(256) void dinv_kernel(const unsigned* __restrict__ deg,
                                                   float* __restrict__ dinv, int N) {
    int i = blockIdx.x * blockDim.x + threadIdx.x;
    if (i < N) {
        float d = (float)(deg[i] + 1u);        // +1 self loop
        dinv[i] = rsqrtf(d);
    }
}

// ---------------------------------------------------------------------------
// agg[i][f] = bias[f]  (aggregation buffer pre-loaded with bias)
// ---------------------------------------------------------------------------
__global__ __launch_bounds__(256) void init_bias(float* __restrict__ agg,
                                                 const float* __restrict__ b, long total) {
    long i = (long)blockIdx.x * blockDim.x + threadIdx.x;
    if (i < total) agg[i] = b[i & (HID - 1)];
}

// ---------------------------------------------------------------------------
// Transpose + zero-pad weights: W[K][HID] -> Wt[HID][KP]  (Wt[col][k])
// Padding rows (k >= K) are zero -> GEMM K-tail is exact with no guards.
// ---------------------------------------------------------------------------
__global__ __launch_bounds__(256) void transpose_pad(const float* __restrict__ W,
                                                     float* __restrict__ Wt,
                                                     int K, int KP) {
    int idx = blockIdx.x * blockDim.x + threadIdx.x;
    int total = HID * KP;
    if (idx < total) {
        int col = idx / KP;
        int k   = idx - col * KP;
        Wt[idx] = (k < K) ? W[(long)k * HID + col] : 0.0f;
    }
}

// ---------------------------------------------------------------------------
// FP32 WMMA GEMM: C[M x 128] = A[M x K] @ W[K x 128]
// One wave computes a 16-row x 128-col strip via 8x v_wmma_f32_16x16x4_f32
// per K-step. Main loop is branch-free and clamp-free (pure pointer
// arithmetic); the K%4 tail is one separate step whose A address is clamped
// and whose B rows beyond K are zero in the padded Wt.
// ---------------------------------------------------------------------------
template <int K, int KP>
__global__ __launch_bounds__(256) void gemm_wmma(const float* __restrict__ A,
                                                 const float* __restrict__ Wt,
                                                 float* __restrict__ C, int M) {
    static_assert((K % 2) == 0 && (KP % 4) == 0 && KP >= K, "bad K");
    constexpr int KMAIN = K & ~3;              // 516 / 128
    const int lane = threadIdx.x & 31;
    const int wave = threadIdx.x >> 5;         // 0..7 (8 waves / block)
    const int half = lane >> 4;                // lane half selects K+0 / K+2
    const int l    = lane & 15;

    const int rowBase = blockIdx.x * 128 + wave * 16;
    const int rowA    = rowBase + l;
    const int rowL    = (rowA < M) ? rowA : (M - 1);   // clamp loads; stores guarded
    const float* aPtr = A + (long)rowL * K + 2 * half;
    const float* wPtr = Wt + (long)l * KP + 2 * half;  // + c*16*KP per column tile

    v8f acc[8] = {};

#pragma unroll 2
    for (int k0 = 0; k0 < KMAIN; k0 += 4) {
        const v2f a = *(const v2f*)(aPtr + k0);
        v2f b[8];
#pragma unroll
        for (int c = 0; c < 8; ++c)
            b[c] = *(const v2f*)(wPtr + (long)c * 16 * KP + k0);
#pragma unroll
        for (int c = 0; c < 8; ++c)
            acc[c] = __builtin_amdgcn_wmma_f32_16x16x4_f32(
                false, a, false, b[c], (short)0, acc[c], false, false);
    }

    if constexpr (KMAIN < K) {
        // single tail K-step: ka in {KMAIN, KMAIN+2}; clamp A address for the
        // half whose ka >= K (its B fragment is the zero padding of Wt)
        const int ka  = KMAIN + 2 * half;
        const int kaC = (ka <= K - 2) ? ka : (K - 2);
        const v2f a = *(const v2f*)(A + (long)rowL * K + kaC);
        v2f b[8];
#pragma unroll
        for (int c = 0; c < 8; ++c)
            b[c] = *(const v2f*)(wPtr + (long)c * 16 * KP + KMAIN);
#pragma unroll
        for (int c = 0; c < 8; ++c)
            acc[c] = __builtin_amdgcn_wmma_f32_16x16x4_f32(
                false, a, false, b[c], (short)0, acc[c], false, false);
    }

    // C/D layout: VGPR r -> row (r + 8*half), lanes stripe N
    float* cBase = C + (long)(rowBase + 8 * half) * HID + l;
    if (rowBase + 16 <= M) {
        // fast path: whole 16-row tile in range, straight-line stores
#pragma unroll
        for (int c = 0; c < 8; ++c) {
#pragma unroll
            for (int r = 0; r < 8; ++r)
                cBase[(long)r * HID + c * 16] = acc[c][r];
        }
    } else {
#pragma unroll
        for (int c = 0; c < 8; ++c) {
#pragma unroll
            for (int r = 0; r < 8; ++r) {
                int row = rowBase + r + 8 * half;
                if (row < M) C[(long)row * HID + c * 16 + l] = acc[c][r];
            }
        }
    }
}

// ---------------------------------------------------------------------------
// Edge aggregation: agg[dst] += dinv[src]*dinv[dst] * h[src]
// one wave per edge; lane handles 4 features (coalesced float4 gather)
// self loops appended logically at e in [E, E+N)
// ---------------------------------------------------------------------------
__global__ __launch_bounds__(256) void scatter_edges(const long long* __restrict__ ei,
                                                     const float* __restrict__ dinv,
                                                     const float* __restrict__ h,
                                                     float* __restrict__ agg,
                                                     int E, int N, int totalWaves) {
    const int w    = (blockIdx.x * blockDim.x + threadIdx.x) >> 5;
    const int lane = threadIdx.x & 31;
    const int T    = E + N;
    for (int e = w; e < T; e += totalWaves) {
        int s, d;
        if (e < E) {
            s = (int)ei[e];
            d = (int)ei[(long)E + e];
        } else {
            s = e - E; d = s;                  // self loop
        }
        const float norm = dinv[s] * dinv[d];
        const float4 v = *(const float4*)(h + (long)s * HID + lane * 4);
        float* dst = agg + (long)d * HID + lane * 4;
        atomicAdd(dst + 0, v.x * norm);
        atomicAdd(dst + 1, v.y * norm);
        atomicAdd(dst + 2, v.z * norm);
        atomicAdd(dst + 3, v.w * norm);
    }
}

// ---------------------------------------------------------------------------
// MLP head: relu -> 128x64 -> relu -> 64x4 ; weights staged in LDS
// ---------------------------------------------------------------------------
__global__ __launch_bounds__(256) void mlp_head(const float* __restrict__ hin,
                                                const float* __restrict__ wm1,
                                                const float* __restrict__ bm1,
                                                const float* __restrict__ wm2,
                                                const float* __restrict__ bm2,
                                                float* __restrict__ out, int N) {
    __shared__ float sw1[HID * MLPH];
    __shared__ float sw2[MLPH * OUTD];
    __shared__ float sb1[MLPH];
    __shared__ float sb2[OUTD];
    for (int i = threadIdx.x; i < HID * MLPH; i += blockDim.x) sw1[i] = wm1[i];
    for (int i = threadIdx.x; i < MLPH * OUTD; i += blockDim.x) sw2[i] = wm2[i];
    if (threadIdx.x < MLPH) sb1[threadIdx.x] = bm1[threadIdx.x];
    if (threadIdx.x < OUTD) sb2[threadIdx.x] = bm2[threadIdx.x];
    __syncthreads();

    const int n = blockIdx.x * blockDim.x + threadIdx.x;
    if (n >= N) return;

    float t[MLPH];
#pragma unroll
    for (int j = 0; j < MLPH; ++j) t[j] = sb1[j];

    const float* row = hin + (long)n * HID;
    for (int k = 0; k < HID; ++k) {
        const float xv = fmaxf(row[k], 0.0f);  // relu on GCN output
#pragma unroll
        for (int j = 0; j < MLPH; ++j) t[j] = fmaf(xv, sw1[k * MLPH + j], t[j]);
    }

    float o[OUTD];
#pragma unroll
    for (int c = 0; c < OUTD; ++c) o[c] = sb2[c];
#pragma unroll
    for (int j = 0; j < MLPH; ++j) {
        const float tj = fmaxf(t[j], 0.0f);
#pragma unroll
        for (int c = 0; c < OUTD; ++c) o[c] = fmaf(tj, sw2[j * OUTD + c], o[c]);
    }
#pragma unroll
    for (int c = 0; c < OUTD; ++c) out[(long)n * OUTD + c] = o[c];
}

// ---------------------------------------------------------------------------
extern "C" void kernel_launch(void* const* d_in, const int* in_sizes, int n_in,
                              void* d_out, int out_size, void* d_ws, size_t ws_size,
                              hipStream_t stream) {
    const float*     x   = (const float*)d_in[0];
    const long long* ei  = (const long long*)d_in[1];   // int64 [2,E]
    const float*     w1  = (const float*)d_in[2];
    const float*     b1  = (const float*)d_in[3];
    const float*     w2  = (const float*)d_in[4];
    const float*     b2  = (const float*)d_in[5];
    const float*     wm1 = (const float*)d_in[6];
    const float*     bm1 = (const float*)d_in[7];
    const float*     wm2 = (const float*)d_in[8];
    const float*     bm2 = (const float*)d_in[9];
    float*           out = (float*)d_out;

    const int N = NNODES, E = NEDGES;
    constexpr int KP1 = (INDIM + 3) & ~3;   // 520
    constexpr int KP2 = HID;                // 128

    char* ws = (char*)d_ws;
    unsigned* deg  = (unsigned*)ws;  ws += ((size_t)N * 4 + 255) & ~(size_t)255;
    float*    dinv = (float*)ws;     ws += ((size_t)N * 4 + 255) & ~(size_t)255;
    float*    wt   = (float*)ws;     ws += ((size_t)HID * KP1 * 4 + 255) & ~(size_t)255;
    float*    h    = (float*)ws;     ws += (size_t)N * HID * 4;
    float*    agg  = (float*)ws;     // reused for both layers

    const long totalFeat  = (long)N * HID;
    const int  gemmBlocks = (N + 127) / 128;
    const int  scatBlocks = 2048;
    const int  scatWaves  = scatBlocks * 256 / 32;

    // degrees + d^{-1/2}
    hipMemsetAsync(deg, 0, (size_t)N * sizeof(unsigned), stream);
    deg_count<<<1024, 256, 0, stream>>>(ei, deg, E);
    dinv_kernel<<<(N + 255) / 256, 256, 0, stream>>>(deg, dinv, N);

    // ---- GCN layer 1 ----
    transpose_pad<<<(HID * KP1 + 255) / 256, 256, 0, stream>>>(w1, wt, INDIM, KP1);
    gemm_wmma<INDIM, KP1><<<gemmBlocks, 256, 0, stream>>>(x, wt, h, N);
    init_bias<<<(int)((totalFeat + 255) / 256), 256, 0, stream>>>(agg, b1, totalFeat);
    scatter_edges<<<scatBlocks, 256, 0, stream>>>(ei, dinv, h, agg, E, N, scatWaves);

    // ---- GCN layer 2 ----
    transpose_pad<<<(HID * KP2 + 255) / 256, 256, 0, stream>>>(w2, wt, HID, KP2);
    gemm_wmma<HID, KP2><<<gemmBlocks, 256, 0, stream>>>(agg, wt, h, N);
    init_bias<<<(int)((totalFeat + 255) / 256), 256, 0, stream>>>(agg, b2, totalFeat);
    scatter_edges<<<scatBlocks, 256, 0, stream>>>(ei, dinv, h, agg, E, N, scatWaves);

    // ---- MLP head ----
    mlp_head<<<(N + 255) / 256, 256, 0, stream>>>(agg, wm1, bm1, wm2, bm2, out, N);
}